// VLLMKVCache_35734127903093
// MI455X (gfx1250) — compile-verified
//
#include <hip/hip_runtime.h>

// ---------------------------------------------------------------------------
// vLLM KV-cache insert: d_out = cache; d_out[bidx[t], boff[t], :, :] = input[t]
// Pure bandwidth problem (~1.07 GiB total traffic -> ~46 us @ 23.3 TB/s).
//   1) 512 MiB streaming copy: B128 + non-temporal, 4x unrolled for MLP.
//   2) 8192 x 4 KiB scatter: gfx1250 async LDS DMA (ASYNCcnt) path.
// ---------------------------------------------------------------------------

typedef float __attribute__((ext_vector_type(4))) f4;  // 16-byte vector

// Bulk copy, 4 independent B128 loads in flight per wave before any store.
__global__ void kv_cache_copy_kernel(const f4* __restrict__ src,
                                     f4* __restrict__ dst,
                                     long long n4) {
    const long long S = (long long)gridDim.x * blockDim.x;   // total threads
    long long i       = (long long)blockIdx.x * blockDim.x + threadIdx.x;

    // Main loop: 4 unit-stride-coalesced NT loads, then 4 NT stores.
    for (; i + 3 * S < n4; i += 4 * S) {
        f4 a = __builtin_nontemporal_load(src + i);          // global_load_b128 NT
        f4 b = __builtin_nontemporal_load(src + i + S);
        f4 c = __builtin_nontemporal_load(src + i + 2 * S);
        f4 d = __builtin_nontemporal_load(src + i + 3 * S);
        __builtin_nontemporal_store(a, dst + i);             // global_store_b128 NT
        __builtin_nontemporal_store(b, dst + i + S);
        __builtin_nontemporal_store(c, dst + i + 2 * S);
        __builtin_nontemporal_store(d, dst + i + 3 * S);
    }
    // Remainder (not taken for the reference shapes; kept for generality).
    for (; i < n4; i += S) {
        f4 v = __builtin_nontemporal_load(src + i);
        __builtin_nontemporal_store(v, dst + i);
    }
}

// Scatter: one 256-thread block per token row (256 lanes x 16B = 4 KiB row).
// gfx1250 async LDS DMA path: global -> LDS -> global, tracked with ASYNCcnt.
// Each wave's lanes cover a contiguous 512B LDS slice, so the only sync needed
// is the per-wave s_wait_asynccnt between the load and the store; s_endpgm's
// implicit wait-idle covers the trailing async store.
__global__ void kv_cache_scatter_kernel(const f4* __restrict__ input,
                                        const int* __restrict__ block_indices,
                                        const int* __restrict__ block_offset,
                                        f4* __restrict__ out,
                                        int row_f4, int block_size) {
    const int t   = blockIdx.x;
    const int tid = threadIdx.x;  // 0..row_f4-1 (256)

    const long long slot = (long long)block_indices[t] * block_size + block_offset[t];
    const f4* src = input + (long long)t * row_f4;
    f4*       dst = out   + slot * row_f4;

    __shared__ __attribute__((aligned(16))) f4 lbuf[256];

    // LDS byte offset of this lane's 16B slot (flat addr low 32 bits = LDS offset).
    unsigned lds_off = (unsigned)(unsigned long long)(&lbuf[tid]);

    // DMA 16B/lane from global into LDS (ASYNCcnt++).
    unsigned long long gsrc = (unsigned long long)(const void*)(src + tid);
    asm volatile("global_load_async_to_lds_b128 %0, %1, off"
                 :: "v"(lds_off), "v"(gsrc) : "memory");

    // Wait for this wave's async load to land in LDS.
    asm volatile("s_wait_asynccnt 0" ::: "memory");

    // DMA 16B/lane from LDS out to the scattered cache slot (ASYNCcnt++).
    unsigned long long gdst = (unsigned long long)(void*)(dst + tid);
    asm volatile("global_store_async_from_lds_b128 %0, %1, off"
                 :: "v"(gdst), "v"(lds_off) : "memory");
}

extern "C" void kernel_launch(void* const* d_in, const int* in_sizes, int n_in,
                              void* d_out, int out_size, void* d_ws, size_t ws_size,
                              hipStream_t stream) {
    // setup_inputs() order: input(f32), cache(f32), block_indices(int), block_offset(int)
    const f4*  input = (const f4*)d_in[0];
    const f4*  cache = (const f4*)d_in[1];
    const int* bidx  = (const int*)d_in[2];
    const int* boff  = (const int*)d_in[3];
    f4*        out   = (f4*)d_out;

    const long long n4         = (long long)in_sizes[1] / 4;    // cache f4 elements (32M)
    const int       num_tokens = in_sizes[2];                   // 8192
    const int       row_floats = in_sizes[0] / num_tokens;      // 8*128 = 1024
    const int       row_f4     = row_floats / 4;                // 256
    const int       BLOCK_SIZE = 128;                           // reference constant

    // 1) Stream the whole cache into d_out: 4 B128 elements per thread,
    //    32768 blocks x 256 threads fills n4 = 4 * total_threads exactly.
    const int copy_threads = 256;
    long long copy_blocks_ll = (n4 + (long long)copy_threads * 4 - 1) /
                               ((long long)copy_threads * 4);
    if (copy_blocks_ll < 1) copy_blocks_ll = 1;
    if (copy_blocks_ll > 1048576) copy_blocks_ll = 1048576;
    kv_cache_copy_kernel<<<(unsigned)copy_blocks_ll, copy_threads, 0, stream>>>(
        cache, out, n4);

    // 2) Scatter token rows over the copy (same stream => ordered after copy).
    kv_cache_scatter_kernel<<<num_tokens, row_f4, 0, stream>>>(
        input, bidx, boff, out, row_f4, BLOCK_SIZE);
}